// SelfAttnPerDimPooler_20100446945824
// MI455X (gfx1250) — compile-verified
//
#include <hip/hip_runtime.h>
#include <math.h>
#include <stdint.h>

typedef float v2f __attribute__((ext_vector_type(2)));
typedef float v8f __attribute__((ext_vector_type(8)));

#define NEGV (-1e9f)

// LDS row strides (floats). Multiples of 4 (16B-aligned rows) with
// (stride/4) coprime to 16 -> column reads across 16 lanes hit 16 distinct banks.
#define XS_S  20   // x / attn@v / attn_out buffer [64][20] (all overlaid)
#define QKV_S 52   // qkv buffer                   [64][52]
#define S_S   68   // logits/attn buffer           [64][68]
#define W_S   20   // weight rows

#if defined(__AMDGCN__)
#define USE_ASYNC_GATHER 1
#else
#define USE_ASYNC_GATHER 0
#endif

static __device__ __forceinline__ v8f wmma4(v2f a, v2f b, v8f c) {
  // V_WMMA_F32_16X16X4_F32: D = A(16x4,f32) * B(4x16,f32) + C(16x16,f32)
  return __builtin_amdgcn_wmma_f32_16x16x4_f32(
      /*neg_a=*/false, a, /*neg_b=*/false, b,
      /*c_mod=*/(short)0, c, /*reuse_a=*/false, /*reuse_b=*/false);
}

// One wave (32 threads) per (batch row, pool). Whole attention-pool in LDS.
__global__ __launch_bounds__(32) void attn_pool_kernel(
    const float* __restrict__ emb,      // [N_SUBJECTS,16]
    const float* __restrict__ w_in_g,   // [48,16]
    const float* __restrict__ b_in_g,   // [48]
    const float* __restrict__ w_out_g,  // [16,16]
    const float* __restrict__ b_out_g,  // [16]
    const float* __restrict__ aw_g,     // [16]
    const float* __restrict__ ab_g,     // [16]
    const int*   __restrict__ fav,      // [B,L]
    const int*   __restrict__ book,     // [B,L]
    float* __restrict__ uemb,           // [B,16]
    float* __restrict__ iemb,           // [B,16]
    int B, int L)
{
  __shared__ __align__(16) float sm[10464];
  float* xs     = sm;            // [64][20]  x tile; later attn@v out; later attn_out
  float* qkv    = sm + 1280;     // [64][52]  q|k|v
  float* s      = sm + 4608;     // [64][68]  logits / attn probs
  float* win    = sm + 8960;     // [48][20]
  float* wout   = sm + 9920;     // [16][20]
  float* bin    = sm + 10240;    // [48]
  float* bout   = sm + 10288;    // [16]
  float* aw     = sm + 10304;    // [16]
  float* ab     = sm + 10320;    // [16]
  float* kmask  = sm + 10336;    // [64]  0.0 if valid token else NEGV
  float* scores = sm + 10400;    // [64]
  float* o      = xs;            // alias: attn@v output (cols 0..15, both heads)
  float* ao     = xs;            // alias: attn_out overwrites o tile-by-tile

  const int lane = threadIdx.x;
  const int half = lane >> 4;    // which K-pair / row-half this lane owns
  const int lr   = lane & 15;    // row (A) or column (B/C) within tile
  const int bid  = blockIdx.x;
  const int pool = blockIdx.y;
  const int* ind = (pool ? book : fav) + (size_t)bid * L;
  float* outp    = (pool ? iemb : uemb) + (size_t)bid * 16;

  // ---- preload weights into LDS ----
  for (int t = lane; t < 48 * 16; t += 32) win[(t >> 4) * W_S + (t & 15)] = w_in_g[t];
  for (int t = lane; t < 16 * 16; t += 32) wout[(t >> 4) * W_S + (t & 15)] = w_out_g[t];
  for (int t = lane; t < 48; t += 32) bin[t] = b_in_g[t];
  if (lane < 16) { bout[lane] = b_out_g[lane]; aw[lane] = aw_g[lane]; ab[lane] = ab_g[lane]; }

  // ---- gather x = subject_emb[indices], build mask, zero-pad rows L..63 ----
  for (int t = lane; t < 64; t += 32) {
    float m = NEGV;
    if (t < L) {
      int idx = ind[t];
      if (idx != 0) m = 0.f;
#if USE_ASYNC_GATHER
      // Per-lane gather straight into LDS (ASYNCcnt-tracked, no VGPR round-trip).
      // The immediate offset is added to both the LDS address and the global address.
      unsigned lds_addr = (unsigned)(uintptr_t)(xs + t * XS_S);   // low 32 bits of
      unsigned long long ga =                                      // flat LDS ptr ==
          (unsigned long long)(uintptr_t)(emb + (size_t)idx * 16); // LDS byte offset
      asm volatile(
          "global_load_async_to_lds_b128 %0, %1, off\n\t"
          "global_load_async_to_lds_b128 %0, %1, off offset:16\n\t"
          "global_load_async_to_lds_b128 %0, %1, off offset:32\n\t"
          "global_load_async_to_lds_b128 %0, %1, off offset:48"
          :: "v"(lds_addr), "v"(ga) : "memory");
#else
      const float4* src = (const float4*)(emb + (size_t)idx * 16);
      float4* dst = (float4*)(xs + t * XS_S);
      dst[0] = src[0]; dst[1] = src[1]; dst[2] = src[2]; dst[3] = src[3];
#endif
    } else {
      float4 z = make_float4(0.f, 0.f, 0.f, 0.f);
      float4* dst = (float4*)(xs + t * XS_S);
      dst[0] = z; dst[1] = z; dst[2] = z; dst[3] = z;
    }
    kmask[t] = m;
  }
#if USE_ASYNC_GATHER
  asm volatile("s_wait_asynccnt 0x0" ::: "memory");
#endif
  __syncthreads();

  // ---- qkv[64,48] = x[64,16] @ Win^T[16,48] + bin ----
  for (int mt = 0; mt < 4; ++mt)
    for (int nb = 0; nb < 3; ++nb) {
      v8f c = {};
      #pragma unroll
      for (int kt = 0; kt < 4; ++kt) {
        int k0 = kt * 4 + half * 2;
        v2f a, bf;
        a.x  = xs[(mt * 16 + lr) * XS_S + k0];
        a.y  = xs[(mt * 16 + lr) * XS_S + k0 + 1];
        bf.x = win[(nb * 16 + lr) * W_S + k0];     // B[k][n] = Win[n][k]
        bf.y = win[(nb * 16 + lr) * W_S + k0 + 1];
        c = wmma4(a, bf, c);
      }
      float bias = bin[nb * 16 + lr];
      #pragma unroll
      for (int j = 0; j < 8; ++j)
        qkv[(mt * 16 + j + half * 8) * QKV_S + nb * 16 + lr] = c[j] + bias;
    }
  __syncthreads();

  const float inv_sqrt_dh = 0.35355339059327373f;  // 1/sqrt(8)
  const int kvmax = (L + 3) >> 2;                  // k-steps covering valid keys

  for (int h = 0; h < 2; ++h) {
    // ---- logits[64,64] = q_h @ k_h^T * inv_sqrt_dh + key_mask ----
    for (int mt = 0; mt < 4; ++mt)
      for (int nt = 0; nt < 4; ++nt) {
        v8f c = {};
        #pragma unroll
        for (int kt = 0; kt < 2; ++kt) {          // dh = 8 -> 2 k-steps
          int k0 = kt * 4 + half * 2;
          v2f a, bf;
          a.x  = qkv[(mt * 16 + lr) * QKV_S + h * 8 + k0];          // q
          a.y  = qkv[(mt * 16 + lr) * QKV_S + h * 8 + k0 + 1];
          bf.x = qkv[(nt * 16 + lr) * QKV_S + 16 + h * 8 + k0];     // k (B[kk][n]=k[n][kk])
          bf.y = qkv[(nt * 16 + lr) * QKV_S + 16 + h * 8 + k0 + 1];
          c = wmma4(a, bf, c);
        }
        float kc = kmask[nt * 16 + lr];
        #pragma unroll
        for (int j = 0; j < 8; ++j)
          s[(mt * 16 + j + half * 8) * S_S + nt * 16 + lr] = c[j] * inv_sqrt_dh + kc;
      }
    __syncthreads();

    // ---- row softmax over all 64 cols (masked/padded cols -> exp(NEG) == 0) ----
    for (int r = lane; r < 64; r += 32) {
      float* row = s + r * S_S;
      float m = row[0];
      for (int j = 1; j < 64; ++j) m = fmaxf(m, row[j]);
      float sum = 0.f;
      for (int j = 0; j < 64; ++j) { float e = expf(row[j] - m); row[j] = e; sum += e; }
      float rinv = 1.f / sum;
      for (int j = 0; j < 64; ++j) row[j] *= rinv;
    }
    __syncthreads();

    // ---- o_h[64,8] = attn[64,:] @ v_h[:,8]  (cols >= 4*kvmax have attn == 0) ----
    for (int mt = 0; mt < 4; ++mt) {
      v8f c = {};
      for (int kt = 0; kt < kvmax; ++kt) {
        int k0 = kt * 4 + half * 2;
        v2f a, bf;
        a.x  = s[(mt * 16 + lr) * S_S + k0];
        a.y  = s[(mt * 16 + lr) * S_S + k0 + 1];
        bf.x = (lr < 8) ? qkv[(k0    ) * QKV_S + 32 + h * 8 + lr] : 0.f;  // v[k][n]
        bf.y = (lr < 8) ? qkv[(k0 + 1) * QKV_S + 32 + h * 8 + lr] : 0.f;
        c = wmma4(a, bf, c);
      }
      if (lr < 8) {
        #pragma unroll
        for (int j = 0; j < 8; ++j)
          o[(mt * 16 + j + half * 8) * XS_S + h * 8 + lr] = c[j];
      }
    }
    __syncthreads();
  }

  // ---- attn_out[64,16] = o @ Wout^T + bout (tile-wise in-place over o) ----
  for (int mt = 0; mt < 4; ++mt) {
    v8f c = {};
    #pragma unroll
    for (int kt = 0; kt < 4; ++kt) {
      int k0 = kt * 4 + half * 2;
      v2f a, bf;
      a.x  = o[(mt * 16 + lr) * XS_S + k0];
      a.y  = o[(mt * 16 + lr) * XS_S + k0 + 1];
      bf.x = wout[lr * W_S + k0];                 // B[k][n] = Wout[n][k]
      bf.y = wout[lr * W_S + k0 + 1];
      c = wmma4(a, bf, c);
    }
    float bias = bout[lr];
    #pragma unroll
    for (int j = 0; j < 8; ++j)
      ao[(mt * 16 + j + half * 8) * XS_S + lr] = c[j] + bias;
  }
  __syncthreads();

  // ---- per-token score, masked softmax over L, weighted sum -> [16] ----
  for (int r = lane; r < 64; r += 32) {
    float sc = 0.f;
    for (int d = 0; d < 16; ++d) sc += ao[r * XS_S + d] * aw[d] + ab[d];
    scores[r] = (kmask[r] == 0.f) ? sc : NEGV;    // reference replaces, not adds
  }
  __syncthreads();
  float m = scores[0];
  for (int r = 1; r < 64; ++r) m = fmaxf(m, scores[r]);  // uniform across lanes
  for (int r = lane; r < 64; r += 32) scores[r] = expf(scores[r] - m);
  __syncthreads();
  float sum = 0.f;
  for (int r = 0; r < 64; ++r) sum += scores[r];
  if (lane < 16) {
    float acc = 0.f;
    for (int r = 0; r < 64; ++r) acc += scores[r] * ao[r * XS_S + lane];
    outp[lane] = acc / sum;
  }
}

__global__ void combine_kernel(const float* __restrict__ uemb,
                               const float* __restrict__ iemb,
                               const int* __restrict__ uidx,
                               const int* __restrict__ iidx,
                               const float* __restrict__ ubias,
                               const float* __restrict__ ibias,
                               const float* __restrict__ gbias,
                               float* __restrict__ out, int B)
{
  int b = blockIdx.x * blockDim.x + threadIdx.x;
  if (b >= B) return;
  float dot = 0.f;
  #pragma unroll
  for (int d = 0; d < 16; ++d) dot += uemb[b * 16 + d] * iemb[b * 16 + d];
  out[b] = dot + ubias[uidx[b]] + ibias[iidx[b]] + gbias[0];
}

extern "C" void kernel_launch(void* const* d_in, const int* in_sizes, int n_in,
                              void* d_out, int out_size, void* d_ws, size_t ws_size,
                              hipStream_t stream) {
  const float* emb   = (const float*)d_in[0];   // subject_emb [100000,16]
  const float* w_in  = (const float*)d_in[1];   // [48,16]
  const float* b_in  = (const float*)d_in[2];   // [48]
  const float* w_out = (const float*)d_in[3];   // [16,16]
  const float* b_out = (const float*)d_in[4];   // [16]
  const float* aw    = (const float*)d_in[5];   // [16]
  const float* ab    = (const float*)d_in[6];   // [16]
  const float* ubias = (const float*)d_in[7];   // [N_USERS,1]
  const float* ibias = (const float*)d_in[8];   // [N_ITEMS,1]
  const float* gbias = (const float*)d_in[9];   // [1]
  const int*   uidx  = (const int*)d_in[10];    // [B]
  const int*   iidx  = (const int*)d_in[11];    // [B]
  const int*   fav   = (const int*)d_in[12];    // [B,L]
  const int*   book  = (const int*)d_in[13];    // [B,L]

  const int B = in_sizes[10];
  const int L = in_sizes[12] / B;               // 50 (must be <= 64)

  float* uemb = (float*)d_ws;                   // [B,16]
  float* iemb = uemb + (size_t)B * 16;          // [B,16]

  dim3 grid(B, 2, 1);                           // y: pool 0 = user, 1 = item
  attn_pool_kernel<<<grid, dim3(32, 1, 1), 0, stream>>>(
      emb, w_in, b_in, w_out, b_out, aw, ab, fav, book, uemb, iemb, B, L);

  const int threads = 256;
  combine_kernel<<<(B + threads - 1) / threads, threads, 0, stream>>>(
      uemb, iemb, uidx, iidx, ubias, ibias, gbias, (float*)d_out, B);
}